// CorticalMap_74672301408410
// MI455X (gfx1250) — compile-verified
//
#include <hip/hip_runtime.h>
#include <hip/hip_bf16.h>
#include <math.h>

// ---------------- problem constants ----------------
#define S       160
#define A       21
#define AA      441          // A*A
#define L       31
#define LL      961          // L*L
#define LP      992          // 31 rows x 32 (padded K space)
#define E       11
#define EE      121          // E*E
#define ITERS   10
#define HOMEO   0.04f
#define AFF_STR 1.5f
#define STRENGTH 1.0f
#define SS      (S*S)        // 25600
#define XDIM    (S + A - 1)  // 180
#define PI_F    3.14159265358979323846f
#define WROW    47           // padded LDS window row width (31 + 15 + 1 pad)

typedef __attribute__((ext_vector_type(2))) float v2f;
typedef __attribute__((ext_vector_type(8))) float v8f;

// ---------------- workspace layout (floats) ----------------
#define WS_ENV   0
#define WS_SRE   (WS_ENV + AA)          // 441
#define WS_LRI   (WS_SRE + EE)          // 562
#define WS_DEN   (WS_LRI + LL)          // 1523
#define WS_LATA  (WS_DEN + SS)          // 27123
#define WS_LATB  (WS_LATA + SS)         // 52723
#define WS_MEAN  (WS_LATB + SS)         // 78323  (total ~104k floats ≈ 416 KB)

// ---------------- d_out layout (floats) ----------------
#define OUT_AFF   0
#define OUT_LAT   SS
#define OUT_CORR  (2*SS)
#define OUT_TILES (2*SS + 1)

// ============================================================
// 1. Envelope init: aff_env (max-normalized), sre (sum-norm),
//    lri (sum-norm). One 1024-thread block; block reductions.
// ============================================================
__global__ void init_env_kernel(float* __restrict__ ws, float* __restrict__ out_corr) {
  __shared__ float red[1024];
  const int tid = threadIdx.x;

  // aff_env: cos(pi*d/21)^2 * circle(10.5); max = 1 at center
  if (tid < AA) {
    float r = (float)(tid / A) - 10.f, c = (float)(tid % A) - 10.f;
    float d = sqrtf(r * r + c * c);
    float v = cosf(PI_F * d / (float)A); v *= v;
    ws[WS_ENV + tid] = (d < (float)A * 0.5f) ? v : 0.f;
  }

  // sre: cos(pi*d/11)^2 * circle(5.5), normalized by sum
  float sval = 0.f;
  if (tid < EE) {
    float r = (float)(tid / E) - 5.f, c = (float)(tid % E) - 5.f;
    float d = sqrtf(r * r + c * c);
    float v = cosf(PI_F * d / (float)E); v *= v;
    sval = (d < (float)E * 0.5f) ? v : 0.f;
  }
  red[tid] = sval; __syncthreads();
  for (int s = 512; s > 0; s >>= 1) { if (tid < s) red[tid] += red[tid + s]; __syncthreads(); }
  float ssum = red[0]; __syncthreads();
  if (tid < EE) ws[WS_SRE + tid] = sval / ssum;

  // lri: cos(pi*d/31)^2 * (1 - inh) * circle(15.5), normalized by sum
  float lval = 0.f;
  if (tid < LL) {
    float r = (float)(tid / L) - 15.f, c = (float)(tid % L) - 15.f;
    float d = sqrtf(r * r + c * c);
    float inh = cosf(PI_F * d / (float)E); inh *= inh;
    inh = (d < (float)E * 0.5f) ? inh : 0.f;
    float base = cosf(PI_F * d / (float)L); base *= base;
    float v = base * (1.f - inh);
    lval = (d < (float)L * 0.5f) ? v : 0.f;
  }
  red[tid] = lval; __syncthreads();
  for (int s = 512; s > 0; s >>= 1) { if (tid < s) red[tid] += red[tid + s]; __syncthreads(); }
  float lsum = red[0]; __syncthreads();
  if (tid < LL) ws[WS_LRI + tid] = lval / lsum;

  if (tid == 0) *out_corr = 0.f;
}

// ============================================================
// 2. Afferent pass: one wave per pixel.
// ============================================================
__global__ void aff_kernel(const float* __restrict__ x,
                           const float* __restrict__ rfs,
                           const float* __restrict__ ada,
                           const float* __restrict__ env,
                           float* __restrict__ out,
                           float* __restrict__ latA,
                           float* __restrict__ lat_mean) {
  const int wave = threadIdx.x >> 5;
  const int lane = threadIdx.x & 31;
  const int l = blockIdx.x * 8 + wave;
  const int y = l / S, xp = l % S;
  float* tiles = out + OUT_TILES + (size_t)l * AA;
  const float* rf = rfs + (size_t)l * AA;
  float sum = 0.f;
  for (int c = lane; c < AA; c += 32) {
    int dy = c / A, dx = c % A;
    float xv = x[(y + dy) * XDIM + (xp + dx)] * env[c];
    tiles[c] = xv;
    sum += xv * fmaxf(rf[c], 0.f);
  }
  for (int m = 16; m >= 1; m >>= 1) sum += __shfl_xor(sum, m, 32);
  if (lane == 0) {
    out[OUT_AFF + l] = sum;
    latA[l] = fmaxf(AFF_STR * sum - ada[l], 0.f);
    lat_mean[l] = 0.f;
  }
}

// ============================================================
// 3. Per-pixel inhibition normalizer: den[l] = sum relu(w)*lri
// ============================================================
__global__ void denom_kernel(const float* __restrict__ latw,
                             const float* __restrict__ lri,
                             float* __restrict__ den) {
  const int wave = threadIdx.x >> 5;
  const int lane = threadIdx.x & 31;
  const int l = blockIdx.x * 8 + wave;
  const float* w = latw + (size_t)l * LL;
  float sum = 0.f;
  for (int c = lane; c < LL; c += 32) sum += fmaxf(w[c], 0.f) * lri[c];
  for (int m = 16; m >= 1; m >>= 1) sum += __shfl_xor(sum, m, 32);
  if (lane == 0) den[l] = sum;
}

// ============================================================
// 4. Shared 11x11 excitatory conv, HOMEO-padded, LDS tile.
// ============================================================
__global__ void conv_kernel(const float* __restrict__ latA,
                            const float* __restrict__ sre,
                            float* __restrict__ latB) {
  __shared__ float tile[26 * 26];
  __shared__ float s_k[EE];
  const int tx = threadIdx.x, ty = threadIdx.y;
  const int tid = ty * 16 + tx;
  if (tid < EE) s_k[tid] = sre[tid];
  const int bx = blockIdx.x * 16, by = blockIdx.y * 16;
  for (int idx = tid; idx < 26 * 26; idx += 256) {
    int r = idx / 26, c = idx % 26;
    int gy = by - 5 + r, gx = bx - 5 + c;
    tile[idx] = (gy >= 0 && gy < S && gx >= 0 && gx < S) ? latA[gy * S + gx] : HOMEO;
  }
  __syncthreads();
  float acc = 0.f;
#pragma unroll
  for (int i = 0; i < E; ++i)
#pragma unroll
    for (int j = 0; j < E; ++j)
      acc += tile[(ty + i) * 26 + (tx + j)] * s_k[i * E + j];
  latB[(by + ty) * S + (bx + tx)] = acc;
}

// ============================================================
// 5. Lateral inhibition via WMMA f32 16x16x4 diagonal trick.
//    K space re-permuted to 31 rows x 32 (padded): dy = kk>>5,
//    dx = kk&31. Padding column is zeroed via s_tab so the hot
//    loop is fully branchless with EXEC = all ones throughout.
//    MODE 0: settle update.  MODE 1: Hebbian correlation scalar.
// ============================================================
template <int MODE>
__global__ void latneg_kernel(const float* __restrict__ src,      // latB (0) or lat_mean (1)
                              const float* __restrict__ latw,
                              const float* __restrict__ lri,
                              const float* __restrict__ denom,
                              const float* __restrict__ aff,      // raw_aff (d_out head)
                              const float* __restrict__ ada,
                              float* __restrict__ latA,
                              float* __restrict__ lat_mean,
                              float* __restrict__ corr) {
  __shared__ float s_tab[LP];            // lri (mode0) / valid mask (mode1), padded layout
  __shared__ float s_win[8][31 * WROW];  // per-wave padded window
  const int tid = threadIdx.x;
  const int wave = tid >> 5, lane = tid & 31;

  // coefficient table in padded-K layout; zeros kill the dummy column
  for (int kk = tid; kk < LP; kk += 256) {
    int dy = kk >> 5, dx = kk & 31;
    float v = 0.f;
    if (dx < 31) v = (MODE == 0) ? lri[dy * 31 + dx] : 1.0f;
    s_tab[kk] = v;
  }

  const int g = blockIdx.x * 8 + wave;          // 1600 groups of 16 pixels
  const int y = g / (S / 16);
  const int x0 = (g % (S / 16)) * 16;

  // stage 31 x 47 padded window (HOMEO halo; col 46 is harmless filler)
  for (int idx = lane; idx < 31 * WROW; idx += 32) {
    int r = idx / WROW, c = idx - r * WROW;
    int gy = y - 15 + r, gx = x0 - 15 + c;
    s_win[wave][idx] = (gy >= 0 && gy < S && gx >= 0 && gx < S) ? src[gy * S + gx] : HOMEO;
  }
  __syncthreads();

  const int m = lane & 15;                      // A row / B col index
  const int koff = (lane >> 4) * 2;             // lanes 16-31 carry K+2,K+3
  const int l0 = y * S + x0;
  const float* wrow = latw + (size_t)(l0 + m) * LL;
  const float* wp = &s_win[wave][0];

  v8f acc = {0.f, 0.f, 0.f, 0.f, 0.f, 0.f, 0.f, 0.f};
#pragma unroll 2
  for (int kb = 0; kb < LP; kb += 4) {
    int kk0 = kb + koff;                        // even
    int dy  = kk0 >> 5;
    int dx0 = kk0 & 31;                         // <= 30 (even)
    int dx1 = dx0 + 1;                          // <= 31 (31 => dummy)
    int dx1c = (dx1 > 30) ? 30 : dx1;           // clamp weight index (value killed by s_tab=0)
    const float* wr = wp + dy * WROW + m;
    int crow = dy * 31;
    v2f a, b;
    a.x = wr[dx0];
    a.y = wr[dx1];
    b.x = fmaxf(wrow[crow + dx0], 0.f) * s_tab[kk0];
    b.y = fmaxf(wrow[crow + dx1c], 0.f) * s_tab[kk0 + 1];
    // D = A(16x4) x B(4x16) + C ; accumulate full 16x16, use diagonal
    acc = __builtin_amdgcn_wmma_f32_16x16x4_f32(false, a, false, b, (short)0, acc,
                                                false, false);
  }

  // diagonal: pixels 0-7 -> lanes 0-7 comp=lane ; pixels 8-15 -> lanes 24-31 comp=lane-24
  const bool has = (lane < 8) || (lane >= 24);
  const int mpix = (lane < 8) ? lane : (lane - 16);
  const int comp = (lane < 8) ? lane : (lane - 24);
  if (has) {
    float s;
    switch (comp) {
      case 0: s = acc[0]; break; case 1: s = acc[1]; break;
      case 2: s = acc[2]; break; case 3: s = acc[3]; break;
      case 4: s = acc[4]; break; case 5: s = acc[5]; break;
      case 6: s = acc[6]; break; default: s = acc[7]; break;
    }
    const int l = l0 + mpix;
    if (MODE == 0) {
      float lneg = s / denom[l];
      float lb = src[l];
      float v = fmaxf((lb - lneg) * STRENGTH + AFF_STR * aff[l] - ada[l], 0.f);
      v = tanhf(v);
      latA[l] = v;
      lat_mean[l] += v;
    } else {
      atomicAdd(corr, s * src[l]);   // src = lat_mean (already /ITERS)
    }
  }
}

// ============================================================
// 6. Finalize: emit lat, scale lat_mean by 1/ITERS
// ============================================================
__global__ void finalize_kernel(const float* __restrict__ latA,
                                float* __restrict__ lat_mean,
                                float* __restrict__ out_lat) {
  const int l = blockIdx.x * 256 + threadIdx.x;
  if (l < SS) {
    out_lat[l] = latA[l];
    lat_mean[l] *= (1.f / (float)ITERS);
  }
}

// ============================================================
extern "C" void kernel_launch(void* const* d_in, const int* in_sizes, int n_in,
                              void* d_out, int out_size, void* d_ws, size_t ws_size,
                              hipStream_t stream) {
  (void)in_sizes; (void)n_in; (void)out_size; (void)ws_size;
  const float* x    = (const float*)d_in[0];
  const float* rfs  = (const float*)d_in[1];
  const float* latw = (const float*)d_in[2];
  const float* ada  = (const float*)d_in[3];
  float* out = (float*)d_out;
  float* ws  = (float*)d_ws;

  float* env  = ws + WS_ENV;
  float* sre  = ws + WS_SRE;
  float* lri  = ws + WS_LRI;
  float* den  = ws + WS_DEN;
  float* latA = ws + WS_LATA;
  float* latB = ws + WS_LATB;
  float* mean = ws + WS_MEAN;

  init_env_kernel<<<1, 1024, 0, stream>>>(ws, out + OUT_CORR);
  aff_kernel<<<SS / 8, 256, 0, stream>>>(x, rfs, ada, env, out, latA, mean);
  denom_kernel<<<SS / 8, 256, 0, stream>>>(latw, lri, den);

  for (int it = 0; it < ITERS; ++it) {
    conv_kernel<<<dim3(10, 10), dim3(16, 16), 0, stream>>>(latA, sre, latB);
    latneg_kernel<0><<<200, 256, 0, stream>>>(latB, latw, lri, den,
                                              out + OUT_AFF, ada, latA, mean, nullptr);
  }

  finalize_kernel<<<100, 256, 0, stream>>>(latA, mean, out + OUT_LAT);
  latneg_kernel<1><<<200, 256, 0, stream>>>(mean, latw, lri, den,
                                            out + OUT_AFF, ada, latA, mean,
                                            out + OUT_CORR);
}